// SlotAttentionTranslEquiv_22428319220324
// MI455X (gfx1250) — compile-verified
//
#include <hip/hip_runtime.h>
#include <math.h>

typedef __attribute__((ext_vector_type(2))) float v2f;
typedef __attribute__((ext_vector_type(8))) float v8f;

#define Bc   8
#define Nc   4096
#define Qc   11
#define QP   16
#define Dc   64
#define Sc   128
#define KV   128
#define HIDc 256
#define NCHUNK 8

// ---------------------------------------------------------------- setup
__global__ void k_setup(const float* __restrict__ slots_in,
                        float* __restrict__ slots, float* __restrict__ pos) {
  int i = blockIdx.x * blockDim.x + threadIdx.x;
  if (i < Bc * Qc * Sc) {
    int t = i % Sc, bq = i / Sc;
    slots[i] = slots_in[bq * (Sc + 2) + t];
  }
  if (i < Bc * Qc * 2) pos[i] = 0.f;   // reference zeroes positions
}

// ------------------------------------------------- LayerNorm of features
__global__ void k_ln_feat(const float* __restrict__ inputs,
                          const float* __restrict__ sc, const float* __restrict__ bi,
                          float* __restrict__ featln, float* __restrict__ gridb) {
  int row = blockIdx.x;            // B*N rows
  int t = threadIdx.x;             // 64 threads
  const float* rp = inputs + (size_t)row * (Dc + 2);
  float x = rp[t];
  __shared__ float red[64];
  red[t] = x; __syncthreads();
  for (int s = 32; s > 0; s >>= 1) { if (t < s) red[t] += red[t + s]; __syncthreads(); }
  float m = red[0] / Dc; __syncthreads();
  float d0 = x - m;
  red[t] = d0 * d0; __syncthreads();
  for (int s = 32; s > 0; s >>= 1) { if (t < s) red[t] += red[t + s]; __syncthreads(); }
  float v = red[0] / Dc;
  featln[(size_t)row * Dc + t] = d0 * rsqrtf(v + 1e-6f) * sc[t] + bi[t];
  if (t < 2) gridb[row * 2 + t] = rp[Dc + t];
}

// ------------------------------------- K,V projection via f32 WMMA 16x16x4
// A 16x4 f32: lane l -> M=l&15, VGPR0 K=2*(l>>4), VGPR1 K=2*(l>>4)+1 (ISA 7.12.2)
// B 4x16 f32: lane l -> N=l&15, same K split. C/D: lane col=l&15, row=r+8*(l>>4).
__global__ void k_kv(const float* __restrict__ featln,
                     const float* __restrict__ Wk, const float* __restrict__ Wv,
                     float* __restrict__ kmat, float* __restrict__ vmat) {
  int blk = blockIdx.x;
  int ct = blk & 7;  blk >>= 3;     // 8 column tiles of 16
  int nt = blk & 255; blk >>= 8;    // 256 row tiles of 16
  int b  = blk;
  int l = threadIdx.x, lane16 = l & 15, kh = l >> 4;
  int row = nt * 16 + lane16;
  int col = ct * 16 + lane16;
  const float* arow = featln + ((size_t)b * Nc + row) * Dc;
  v8f ck = {}; v8f cv = {};
  for (int k = 0; k < Dc; k += 4) {
    v2f a;  a.x  = arow[k + 2 * kh];              a.y  = arow[k + 2 * kh + 1];
    v2f bk; bk.x = Wk[(k + 2 * kh) * KV + col];   bk.y = Wk[(k + 2 * kh + 1) * KV + col];
    v2f bv; bv.x = Wv[(k + 2 * kh) * KV + col];   bv.y = Wv[(k + 2 * kh + 1) * KV + col];
    ck = __builtin_amdgcn_wmma_f32_16x16x4_f32(false, a, false, bk, (short)0, ck, false, false);
    cv = __builtin_amdgcn_wmma_f32_16x16x4_f32(false, a, false, bv, (short)0, cv, false, false);
  }
  for (int r = 0; r < 8; r++) {
    int orow = nt * 16 + r + 8 * kh;
    kmat[((size_t)b * Nc + orow) * KV + col] = ck[r];
    vmat[((size_t)b * Nc + orow) * KV + col] = cv[r];
  }
}

// ------------------- per-round slot prep: LN, q, w = Wge@q, affine consts
__global__ void k_slotprep(const float* __restrict__ slots, const float* __restrict__ pos,
                           const float* __restrict__ lnq_s, const float* __restrict__ lnq_b,
                           const float* __restrict__ Wq,  const float* __restrict__ Wge,
                           const float* __restrict__ Wgp, const float* __restrict__ bgp,
                           const float* __restrict__ bge,
                           float* __restrict__ wvec, float* __restrict__ uvec) {
  int b = blockIdx.x / QP, q = blockIdx.x % QP;
  int t = threadIdx.x;                         // 128 threads
  if (q >= Qc) {                               // zero the Q-padding columns
    wvec[(b * QP + q) * KV + t] = 0.f;
    if (t < 4) uvec[(b * QP + q) * 4 + t] = 0.f;
    return;
  }
  __shared__ float sh[Sc], qh[Sc], red[Sc];
  float h = slots[((b * Qc) + q) * Sc + t];
  red[t] = h; __syncthreads();
  for (int s = 64; s > 0; s >>= 1) { if (t < s) red[t] += red[t + s]; __syncthreads(); }
  float m = red[0] / Sc; __syncthreads();
  float d = h - m;
  red[t] = d * d; __syncthreads();
  for (int s = 64; s > 0; s >>= 1) { if (t < s) red[t] += red[t + s]; __syncthreads(); }
  float var = red[0] / Sc; __syncthreads();
  sh[t] = d * rsqrtf(var + 1e-6f) * lnq_s[t] + lnq_b[t];
  __syncthreads();
  float acc = 0.f;                              // q-vector
  for (int s2 = 0; s2 < Sc; s2++) acc += sh[s2] * Wq[s2 * KV + t];
  qh[t] = acc * (1.f / sqrtf((float)KV));
  __syncthreads();
  float wa = 0.f;                               // w = Wge @ q
  for (int d2 = 0; d2 < KV; d2++) wa += Wge[t * KV + d2] * qh[d2];
  wvec[(b * QP + q) * KV + t] = wa;
  float p0 = Wgp[t] * wa;
  float p1 = Wgp[KV + t] * wa;
  float pc = bgp[t] * wa + qh[t] * bge[t];
  red[t] = p0; __syncthreads();
  for (int s = 64; s > 0; s >>= 1) { if (t < s) red[t] += red[t + s]; __syncthreads(); }
  float u0 = red[0]; __syncthreads();
  red[t] = p1; __syncthreads();
  for (int s = 64; s > 0; s >>= 1) { if (t < s) red[t] += red[t + s]; __syncthreads(); }
  float u1 = red[0]; __syncthreads();
  red[t] = pc; __syncthreads();
  for (int s = 64; s > 0; s >>= 1) { if (t < s) red[t] += red[t + s]; __syncthreads(); }
  if (t == 0) {
    float px = pos[(b * Qc + q) * 2 + 0], py = pos[(b * Qc + q) * 2 + 1];
    uvec[(b * QP + q) * 4 + 0] = u0;
    uvec[(b * QP + q) * 4 + 1] = u1;
    uvec[(b * QP + q) * 4 + 2] = red[0] - px * u0 - py * u1;  // const term
    uvec[(b * QP + q) * 4 + 3] = 0.f;
  }
}

// ------------------------------ logits = K @ W^T + rank-2 grid term (WMMA)
__global__ void k_logits(const float* __restrict__ kmat, const float* __restrict__ wvec,
                         const float* __restrict__ uvec, const float* __restrict__ gridb,
                         float* __restrict__ attn) {
  int blk = blockIdx.x;
  int nt = blk & 255, b = blk >> 8;
  int l = threadIdx.x, lane16 = l & 15, kh = l >> 4;
  const float* krow = kmat + ((size_t)b * Nc + nt * 16 + lane16) * KV;
  const float* wrow = wvec + (b * QP + lane16) * KV;
  v8f c = {};
  for (int k = 0; k < KV; k += 4) {
    v2f a; a.x = krow[k + 2 * kh]; a.y = krow[k + 2 * kh + 1];
    v2f w; w.x = wrow[k + 2 * kh]; w.y = wrow[k + 2 * kh + 1];
    c = __builtin_amdgcn_wmma_f32_16x16x4_f32(false, a, false, w, (short)0, c, false, false);
  }
  float u0 = uvec[(b * QP + lane16) * 4 + 0];
  float u1 = uvec[(b * QP + lane16) * 4 + 1];
  float cc = uvec[(b * QP + lane16) * 4 + 2];
  for (int r = 0; r < 8; r++) {
    int n = nt * 16 + r + 8 * kh;
    float g0 = gridb[(b * Nc + n) * 2 + 0];
    float g1 = gridb[(b * Nc + n) * 2 + 1];
    attn[((size_t)b * Nc + n) * QP + lane16] = c[r] + g0 * u0 + g1 * u1 + cc;
  }
}

// -------------------------------------------------- softmax over Q (axis=-2)
__global__ void k_softmax(float* __restrict__ attn) {
  int i = blockIdx.x * blockDim.x + threadIdx.x;
  if (i >= Bc * Nc) return;
  float* p = attn + (size_t)i * QP;
  float m = -1e30f;
  for (int q = 0; q < Qc; q++) m = fmaxf(m, p[q]);
  float e[Qc]; float s = 0.f;
  for (int q = 0; q < Qc; q++) { e[q] = expf(p[q] - m); s += e[q]; }
  float inv = 1.f / s;
  for (int q = 0; q < Qc; q++) p[q] = e[q] * inv;
  for (int q = Qc; q < QP; q++) p[q] = 0.f;          // clean padding rows
}

// ------------------- per-(b,q) reductions over N: denom + new positions
__global__ void k_reduce(const float* __restrict__ attn, const float* __restrict__ gridb,
                         float* __restrict__ denom, float* __restrict__ pos) {
  int b = blockIdx.x / Qc, q = blockIdx.x % Qc;
  int t = threadIdx.x;                 // 256 threads
  float s = 0.f, s0 = 0.f, s1 = 0.f;
  for (int n = t; n < Nc; n += 256) {
    float a  = attn[((size_t)b * Nc + n) * QP + q];
    float g0 = gridb[(b * Nc + n) * 2 + 0];
    float g1 = gridb[(b * Nc + n) * 2 + 1];
    s += a; s0 += a * g0; s1 += a * g1;
  }
  __shared__ float r0[256], r1[256], r2[256];
  r0[t] = s; r1[t] = s0; r2[t] = s1; __syncthreads();
  for (int k = 128; k > 0; k >>= 1) {
    if (t < k) { r0[t] += r0[t + k]; r1[t] += r1[t + k]; r2[t] += r2[t + k]; }
    __syncthreads();
  }
  if (t == 0) {
    float inv = 1.f / (r0[0] + 1e-8f);
    denom[b * QP + q] = inv;
    pos[(b * Qc + q) * 2 + 0] = r1[0] * inv;
    pos[(b * Qc + q) * 2 + 1] = r2[0] * inv;
  }
}

__global__ void k_zero(float* __restrict__ p, int n) {
  int i = blockIdx.x * blockDim.x + threadIdx.x;
  if (i < n) p[i] = 0.f;
}

// ----------------------- updates = attn @ V (WMMA, chunked over N + atomics)
__global__ void k_updates(const float* __restrict__ attn, const float* __restrict__ vmat,
                          float* __restrict__ upd) {
  int blk = blockIdx.x;
  int ch = blk & 7; blk >>= 3;       // NCHUNK chunks of N/NCHUNK
  int dt = blk & 7; blk >>= 3;       // 8 d-tiles of 16
  int b  = blk;
  int l = threadIdx.x, lane16 = l & 15, kh = l >> 4;
  int d = dt * 16 + lane16;
  v8f c = {};
  int n0 = ch * (Nc / NCHUNK);
  for (int n = n0; n < n0 + Nc / NCHUNK; n += 4) {
    v2f a;
    a.x = attn[((size_t)b * Nc + n + 2 * kh) * QP + lane16];
    a.y = attn[((size_t)b * Nc + n + 2 * kh + 1) * QP + lane16];
    v2f vv;
    vv.x = vmat[((size_t)b * Nc + n + 2 * kh) * KV + d];
    vv.y = vmat[((size_t)b * Nc + n + 2 * kh + 1) * KV + d];
    c = __builtin_amdgcn_wmma_f32_16x16x4_f32(false, a, false, vv, (short)0, c, false, false);
  }
  for (int r = 0; r < 8; r++) {
    int q = r + 8 * kh;
    atomicAdd(&upd[((size_t)b * QP + q) * KV + d], c[r]);
  }
}

// --------------------------------------------- fused GRU + residual MLP
__global__ void k_gru_mlp(const float* __restrict__ upd, const float* __restrict__ denom,
                          float* __restrict__ slots,
                          const float* __restrict__ Wi, const float* __restrict__ bi,
                          const float* __restrict__ Wh, const float* __restrict__ bhn,
                          const float* __restrict__ mls, const float* __restrict__ mlb,
                          const float* __restrict__ W1, const float* __restrict__ b1,
                          const float* __restrict__ W2, const float* __restrict__ b2) {
  int b = blockIdx.x / Qc, q = blockIdx.x % Qc;
  int t = threadIdx.x;                         // 128 threads
  __shared__ float xs[Sc], hs[Sc], red[Sc], ys[Sc], hid[HIDc];
  float dn = denom[b * QP + q];
  xs[t] = upd[((size_t)b * QP + q) * KV + t] * dn;   // normalized updates
  hs[t] = slots[((b * Qc) + q) * Sc + t];
  __syncthreads();
  float gr = bi[t], gz = bi[Sc + t], gn = bi[2 * Sc + t];
  float hr = 0.f, hz = 0.f, hn = 0.f;
  for (int e = 0; e < Sc; e++) {
    float xe = xs[e], he = hs[e];
    gr += xe * Wi[e * 3 * Sc + t];
    gz += xe * Wi[e * 3 * Sc + Sc + t];
    gn += xe * Wi[e * 3 * Sc + 2 * Sc + t];
    hr += he * Wh[e * 3 * Sc + t];
    hz += he * Wh[e * 3 * Sc + Sc + t];
    hn += he * Wh[e * 3 * Sc + 2 * Sc + t];
  }
  float r = 1.f / (1.f + expf(-(gr + hr)));
  float z = 1.f / (1.f + expf(-(gz + hz)));
  float nn = tanhf(gn + r * (hn + bhn[t]));
  float hnew = (1.f - z) * nn + z * hs[t];
  __syncthreads();
  // MLP with pre-LN and residual
  red[t] = hnew; __syncthreads();
  for (int s = 64; s > 0; s >>= 1) { if (t < s) red[t] += red[t + s]; __syncthreads(); }
  float m = red[0] / Sc; __syncthreads();
  float dd = hnew - m;
  red[t] = dd * dd; __syncthreads();
  for (int s = 64; s > 0; s >>= 1) { if (t < s) red[t] += red[t + s]; __syncthreads(); }
  float var = red[0] / Sc;
  ys[t] = dd * rsqrtf(var + 1e-6f) * mls[t] + mlb[t];
  __syncthreads();
  for (int j = t; j < HIDc; j += Sc) {
    float a = b1[j];
    for (int s2 = 0; s2 < Sc; s2++) a += ys[s2] * W1[s2 * HIDc + j];
    hid[j] = fmaxf(a, 0.f);
  }
  __syncthreads();
  float o = b2[t];
  for (int j = 0; j < HIDc; j++) o += hid[j] * W2[j * Sc + t];
  slots[((b * Qc) + q) * Sc + t] = hnew + o;
}

// ---------------------------------------------------------------- output
__global__ void k_final(const float* __restrict__ slots, const float* __restrict__ pos,
                        float* __restrict__ out) {
  int i = blockIdx.x * blockDim.x + threadIdx.x;
  if (i >= Bc * Qc * (Sc + 2)) return;
  int t = i % (Sc + 2), bq = i / (Sc + 2);
  out[i] = (t < Sc) ? slots[bq * Sc + t] : pos[bq * 2 + (t - Sc)];
}

extern "C" void kernel_launch(void* const* d_in, const int* in_sizes, int n_in,
                              void* d_out, int out_size, void* d_ws, size_t ws_size,
                              hipStream_t stream) {
  const float* slots_in = (const float*)d_in[0];
  const float* inputs   = (const float*)d_in[1];
  const float* ln_s     = (const float*)d_in[2];
  const float* ln_b     = (const float*)d_in[3];
  const float* Wq       = (const float*)d_in[4];
  const float* Wk       = (const float*)d_in[5];
  const float* Wv       = (const float*)d_in[6];
  const float* Wgp      = (const float*)d_in[7];
  const float* bgp      = (const float*)d_in[8];
  const float* Wge      = (const float*)d_in[9];
  const float* bge      = (const float*)d_in[10];
  const float* lnq_s    = (const float*)d_in[11];
  const float* lnq_b    = (const float*)d_in[12];
  const float* gru_Wi   = (const float*)d_in[13];
  const float* gru_bi   = (const float*)d_in[14];
  const float* gru_Wh   = (const float*)d_in[15];
  const float* gru_bhn  = (const float*)d_in[16];
  const float* mlp_ls   = (const float*)d_in[17];
  const float* mlp_lb   = (const float*)d_in[18];
  const float* mlp_W1   = (const float*)d_in[19];
  const float* mlp_b1   = (const float*)d_in[20];
  const float* mlp_W2   = (const float*)d_in[21];
  const float* mlp_b2   = (const float*)d_in[22];

  float* ws = (float*)d_ws;
  size_t off = 0;
  float* featln = ws + off; off += (size_t)Bc * Nc * Dc;
  float* gridb  = ws + off; off += (size_t)Bc * Nc * 2;
  float* kmat   = ws + off; off += (size_t)Bc * Nc * KV;
  float* vmat   = ws + off; off += (size_t)Bc * Nc * KV;
  float* attn   = ws + off; off += (size_t)Bc * Nc * QP;
  float* wvec   = ws + off; off += (size_t)Bc * QP * KV;
  float* uvec   = ws + off; off += (size_t)Bc * QP * 4;
  float* slotsw = ws + off; off += (size_t)Bc * Qc * Sc;
  float* pos    = ws + off; off += (size_t)Bc * Qc * 2;
  float* denom  = ws + off; off += (size_t)Bc * QP;
  float* upd    = ws + off; off += (size_t)Bc * QP * KV;

  k_setup<<<(Bc * Qc * Sc + 255) / 256, 256, 0, stream>>>(slots_in, slotsw, pos);
  k_ln_feat<<<Bc * Nc, 64, 0, stream>>>(inputs, ln_s, ln_b, featln, gridb);
  k_kv<<<Bc * 256 * 8, 32, 0, stream>>>(featln, Wk, Wv, kmat, vmat);

  for (int round = 0; round < 4; round++) {
    k_slotprep<<<Bc * QP, 128, 0, stream>>>(slotsw, pos, lnq_s, lnq_b, Wq, Wge,
                                            Wgp, bgp, bge, wvec, uvec);
    k_logits<<<Bc * 256, 32, 0, stream>>>(kmat, wvec, uvec, gridb, attn);
    k_softmax<<<(Bc * Nc + 255) / 256, 256, 0, stream>>>(attn);
    k_reduce<<<Bc * Qc, 256, 0, stream>>>(attn, gridb, denom, pos);
    if (round < 3) {
      k_zero<<<(Bc * QP * KV + 255) / 256, 256, 0, stream>>>(upd, Bc * QP * KV);
      k_updates<<<Bc * 8 * NCHUNK, 32, 0, stream>>>(attn, vmat, upd);
      k_gru_mlp<<<Bc * Qc, 128, 0, stream>>>(upd, denom, slotsw, gru_Wi, gru_bi,
                                             gru_Wh, gru_bhn, mlp_ls, mlp_lb,
                                             mlp_W1, mlp_b1, mlp_W2, mlp_b2);
    }
  }
  k_final<<<(Bc * Qc * (Sc + 2) + 255) / 256, 256, 0, stream>>>(slotsw, pos, (float*)d_out);
}